// GCN_12352325943364
// MI455X (gfx1250) — compile-verified
//
#include <hip/hip_runtime.h>
#include <math.h>

typedef float v2f __attribute__((ext_vector_type(2)));
typedef float v8f __attribute__((ext_vector_type(8)));

#define N_NODES 50000
#define F_IN    128
#define F_HID   128
#define F_OUT   40

// ---------------------------------------------------------------------------
// Utility kernels
// ---------------------------------------------------------------------------
__global__ void zero_f32(float* __restrict__ p, int n) {
    int i = blockIdx.x * blockDim.x + threadIdx.x;
    if (i < n) p[i] = 0.0f;
}

__global__ void deg_acc(const int* __restrict__ dst, float* __restrict__ deg, int E) {
    int e = blockIdx.x * blockDim.x + threadIdx.x;
    if (e < E) unsafeAtomicAdd(&deg[dst[e]], 1.0f);
}

__global__ void make_dinv(float* __restrict__ deg, int n) {
    int i = blockIdx.x * blockDim.x + threadIdx.x;
    if (i < n) deg[i] = rsqrtf(deg[i] + 1.0f);   // deg includes self-loop
}

// ---------------------------------------------------------------------------
// f32 WMMA GEMM:  H[M x NC] = X[M x 128] * W[128 x NC]
// One wave computes a 16-row strip across NT 16-col tiles, reusing the A frag.
// A frag (16x4 f32, 2 VGPR): lanes 0-15 -> K={k,k+1}, lanes 16-31 -> K={k+2,k+3}
// B frag (4x16 f32, 2 VGPR): mirrored K split across lane halves
// C/D  (16x16 f32, 8 VGPR):  M = r + 8*(lane>=16), N = lane%16
// ---------------------------------------------------------------------------
template <int NT, int NC>
__global__ void gemm_wmma_f32(const float* __restrict__ X,
                              const float* __restrict__ W,
                              float* __restrict__ H, int M) {
    const int lane = threadIdx.x & 31;
    const int wave = threadIdx.x >> 5;
    const int mt   = blockIdx.x * (blockDim.x >> 5) + wave;
    if (mt * 16 >= M) return;                 // wave-uniform: EXEC stays all-1s

    const int half = lane >> 4;               // 0: lanes 0-15, 1: lanes 16-31
    const int l16  = lane & 15;

    v8f c[NT] = {};

    const float* xrow = X + (size_t)(mt * 16 + l16) * 128;

    for (int k = 0; k < 128; k += 4) {
        v2f a;
        a.x = xrow[k + 2 * half];
        a.y = xrow[k + 2 * half + 1];
#pragma unroll
        for (int t = 0; t < NT; ++t) {
            const int n = t * 16 + l16;
            v2f b;
            if ((NC % 16 == 0) || (n < NC)) {
                b.x = W[(size_t)(k + 2 * half) * NC + n];
                b.y = W[(size_t)(k + 2 * half + 1) * NC + n];
            } else {
                b.x = 0.0f; b.y = 0.0f;
            }
            c[t] = __builtin_amdgcn_wmma_f32_16x16x4_f32(
                false, a, false, b, (short)0, c[t], false, false);
        }
    }

#pragma unroll
    for (int t = 0; t < NT; ++t) {
        const int n = t * 16 + l16;
        if ((NC % 16 == 0) || (n < NC)) {
            float* out = H + (size_t)(mt * 16 + half * 8) * NC + n;
#pragma unroll
            for (int r = 0; r < 8; ++r) out[(size_t)r * NC] = c[t][r];
        }
    }
}

// ---------------------------------------------------------------------------
// Edge aggregation: AGG[dst, f] += H[src, f] * dinv[src] * dinv[dst]
// One thread per (edge, feature); f32 hardware atomics hit the L2 atomic units.
// ---------------------------------------------------------------------------
template <int F>
__global__ void agg_edges(const int* __restrict__ src, const int* __restrict__ dst,
                          const float* __restrict__ dinv,
                          const float* __restrict__ H, float* __restrict__ AGG,
                          int E) {
    long long gid = (long long)blockIdx.x * blockDim.x + threadIdx.x;
    if (gid >= (long long)E * F) return;
    int e = (int)(gid / F);
    int f = (int)(gid - (long long)e * F);
    int s = src[e], d = dst[e];
    float norm = dinv[s] * dinv[d];
    unsafeAtomicAdd(&AGG[(size_t)d * F + f], H[(size_t)s * F + f] * norm);
}

// ---------------------------------------------------------------------------
// Layer-1 epilogue: self-loop + bias + ReLU, in place into AGG
// ---------------------------------------------------------------------------
template <int F>
__global__ void finalize_relu(const float* __restrict__ H, const float* __restrict__ b,
                              const float* __restrict__ dinv, float* __restrict__ AGG,
                              int Nn) {
    long long gid = (long long)blockIdx.x * blockDim.x + threadIdx.x;
    if (gid >= (long long)Nn * F) return;
    int i = (int)(gid / F);
    int f = (int)(gid - (long long)i * F);
    float di = dinv[i];
    float v  = AGG[gid] + H[gid] * di * di + b[f];
    AGG[gid] = v > 0.0f ? v : 0.0f;
}

// ---------------------------------------------------------------------------
// Layer-2 epilogue + log_softmax over 40 classes; one wave32 per node.
// Lane l holds f=l (all 32 valid) and f=l+32 (lanes 0-7).
// ---------------------------------------------------------------------------
__global__ void finalize_logsoftmax(const float* __restrict__ H2,
                                    const float* __restrict__ b2,
                                    const float* __restrict__ dinv,
                                    const float* __restrict__ AGG2,
                                    float* __restrict__ OUT, int Nn) {
    const int lane = threadIdx.x & 31;
    const int wave = threadIdx.x >> 5;
    const int node = blockIdx.x * (blockDim.x >> 5) + wave;
    if (node >= Nn) return;

    const float di = dinv[node];
    const float dd = di * di;
    const size_t base = (size_t)node * F_OUT;

    float v0 = AGG2[base + lane] + H2[base + lane] * dd + b2[lane];
    float v1 = -INFINITY;
    if (lane < 8)
        v1 = AGG2[base + 32 + lane] + H2[base + 32 + lane] * dd + b2[32 + lane];

    float m = fmaxf(v0, v1);
#pragma unroll
    for (int off = 16; off > 0; off >>= 1)
        m = fmaxf(m, __shfl_xor(m, off, 32));

    float s = __expf(v0 - m) + ((lane < 8) ? __expf(v1 - m) : 0.0f);
#pragma unroll
    for (int off = 16; off > 0; off >>= 1)
        s += __shfl_xor(s, off, 32);

    const float ls = __logf(s);
    OUT[base + lane] = v0 - m - ls;
    if (lane < 8) OUT[base + 32 + lane] = v1 - m - ls;
}

// ---------------------------------------------------------------------------
// Host-side orchestration (all on `stream`, graph-capture safe)
// ---------------------------------------------------------------------------
extern "C" void kernel_launch(void* const* d_in, const int* in_sizes, int n_in,
                              void* d_out, int out_size, void* d_ws, size_t ws_size,
                              hipStream_t stream) {
    const float* x  = (const float*)d_in[0];   // [50000,128]
    const float* W1 = (const float*)d_in[1];   // [128,128]
    const float* b1 = (const float*)d_in[2];   // [128]
    const float* W2 = (const float*)d_in[3];   // [128,40]
    const float* b2 = (const float*)d_in[4];   // [40]
    const int*   ei = (const int*)d_in[5];     // [2,800000] (JAX x64 off -> int32)
    const int E = in_sizes[5] / 2;
    const int Nn = N_NODES;
    const int* src = ei;
    const int* dst = ei + E;

    float* out = (float*)d_out;                // [50000,40]

    // Workspace layout (floats)
    float* ws   = (float*)d_ws;
    float* dinv = ws;                              // 50048 (padded)
    float* h1   = dinv + 50048;                    // 50000*128
    float* agg1 = h1   + (size_t)Nn * F_HID;       // 50000*128 (becomes h_relu)
    float* h2   = agg1 + (size_t)Nn * F_HID;       // 50000*40
    float* agg2 = h2   + (size_t)Nn * F_OUT;       // 50000*40

    const int TPB = 256;

    // Degrees -> dinv
    zero_f32<<<(Nn + TPB - 1) / TPB, TPB, 0, stream>>>(dinv, Nn);
    deg_acc<<<(E + TPB - 1) / TPB, TPB, 0, stream>>>(dst, dinv, E);
    make_dinv<<<(Nn + TPB - 1) / TPB, TPB, 0, stream>>>(dinv, Nn);

    // Layer 1: h1 = x @ W1   (3125 m-tiles, 8 waves/block)
    {
        int mtiles = Nn / 16;                       // 3125
        int blocks = (mtiles + 7) / 8;              // 391
        gemm_wmma_f32<8, 128><<<blocks, TPB, 0, stream>>>(x, W1, h1, Nn);
    }

    // Aggregate layer 1
    {
        long long total = (long long)Nn * F_HID;
        zero_f32<<<(int)((total + TPB - 1) / TPB), TPB, 0, stream>>>(agg1, (int)total);
        long long work = (long long)E * F_HID;
        agg_edges<F_HID><<<(int)((work + TPB - 1) / TPB), TPB, 0, stream>>>(
            src, dst, dinv, h1, agg1, E);
        finalize_relu<F_HID><<<(int)((total + TPB - 1) / TPB), TPB, 0, stream>>>(
            h1, b1, dinv, agg1, Nn);
    }

    // Layer 2: h2 = relu(h) @ W2  (N=40 -> 3 n-tiles with predication)
    {
        int mtiles = Nn / 16;
        int blocks = (mtiles + 7) / 8;
        gemm_wmma_f32<3, 40><<<blocks, TPB, 0, stream>>>(agg1, W2, h2, Nn);
    }

    // Aggregate layer 2 + log_softmax
    {
        long long total = (long long)Nn * F_OUT;
        zero_f32<<<(int)((total + TPB - 1) / TPB), TPB, 0, stream>>>(agg2, (int)total);
        long long work = (long long)E * F_OUT;
        agg_edges<F_OUT><<<(int)((work + TPB - 1) / TPB), TPB, 0, stream>>>(
            src, dst, dinv, h2, agg2, E);
        int blocks = (Nn + 7) / 8;                  // 8 waves per 256-thread block
        finalize_logsoftmax<<<blocks, TPB, 0, stream>>>(h2, b2, dinv, agg2, out, Nn);
    }
}